// SGConvModel_40578851012868
// MI455X (gfx1250) — compile-verified
//
#include <hip/hip_runtime.h>

// ---------------------------------------------------------------------------
// SGConv x2 (PyG SGConv K=1) for MI455X / gfx1250, wave32.
//
//   deg[v]   = 1 + #incoming edges              (self loop included)
//   dinv[v]  = rsqrt(deg[v])
//   agg[d]  += x[s] * dinv[s]*dinv[d]           (incl. self loop s==d)
//   h        = relu(agg @ W1^T + b1)            (WMMA f32 16x16x4)
//   agg2[d] += h[s]  * dinv[s]*dinv[d]
//   out      = agg2 @ W2^T + b2                 (WMMA f32 16x16x4, 40 cols)
//
// Scatter/gather phase is L2-resident (x/agg ~25.6MB each, L2 = 192MB) ->
// plain global_atomic_add_f32 is the right data path. GEMMs use the fp32
// WMMA so no precision is lost vs the f32 reference.
// ---------------------------------------------------------------------------

#define F_DIM 128

typedef float v2f __attribute__((ext_vector_type(2)));
typedef float v8f __attribute__((ext_vector_type(8)));

// ---- propagation kernels ---------------------------------------------------

__global__ void sgc_init_deg(float* __restrict__ deg, int n) {
    int i = blockIdx.x * blockDim.x + threadIdx.x;
    if (i < n) deg[i] = 1.0f;               // self-loop contribution
}

__global__ void sgc_edge_deg(const int* __restrict__ dst, float* __restrict__ deg, int e) {
    int i = blockIdx.x * blockDim.x + threadIdx.x;
    if (i < e) atomicAdd(&deg[dst[i]], 1.0f);
}

__global__ void sgc_rsqrt(float* __restrict__ deg, int n) {
    int i = blockIdx.x * blockDim.x + threadIdx.x;
    if (i < n) deg[i] = rsqrtf(deg[i]);     // deg >= 1 always (self loop)
}

// agg[v][:] = x[v][:] * dinv[v]^2   (the self-loop term), float4-vectorized.
__global__ void sgc_self_init(const float* __restrict__ X, const float* __restrict__ dinv,
                              float* __restrict__ AGG, int n) {
    long long t = (long long)blockIdx.x * blockDim.x + threadIdx.x;  // n*32 float4s
    if (t >= (long long)n * (F_DIM / 4)) return;
    int v = (int)(t >> 5);
    float w = dinv[v];
    w *= w;
    float4 xv = ((const float4*)X)[t];
    float4 o;
    o.x = xv.x * w; o.y = xv.y * w; o.z = xv.z * w; o.w = xv.w * w;
    ((float4*)AGG)[t] = o;
}

// 128 threads per edge (one feature each): coalesced gather of x[src][:],
// atomic scatter-add into agg[dst][:]. Both buffers are L2-resident.
__global__ void sgc_scatter(const int* __restrict__ src, const int* __restrict__ dst,
                            const float* __restrict__ dinv, const float* __restrict__ X,
                            float* __restrict__ AGG, int e) {
    long long t = (long long)blockIdx.x * blockDim.x + threadIdx.x;  // e*128 threads
    if (t >= (long long)e * F_DIM) return;
    int ed = (int)(t >> 7);
    int f  = (int)(t & (F_DIM - 1));
    int s = src[ed];
    int d = dst[ed];
    float norm = dinv[s] * dinv[d];
    atomicAdd(AGG + (long long)d * F_DIM + f, X[(long long)s * F_DIM + f] * norm);
}

// ---- WMMA GEMM: Y[M x NC] = X[M x 128] @ W[NC x 128]^T + bias (opt. ReLU) --
//
// One wave per 16-row M-tile; each wave carries NT = ceil(NC/16) accumulator
// tiles so the A fragment is loaded once per K-step and reused NT times.
//
// V_WMMA_F32_16X16X4_F32 operand layout (ISA 7.12.2, wave32):
//   A (16x4):  lane lo=lane&15 is row M=lo; half hi=lane>>4 selects K pair:
//              a[j] = A[lo][2*hi + j]
//   B (4x16):  b[j] = B[K = j + 2*hi][N = lo]  (B = W^T -> W[col][k])
//   C/D:       c[r] at lane -> (M = r + 8*hi, N = lo)
template <int NC, bool RELU>
__global__ void __launch_bounds__(128)
sgc_gemm_wmma(const float* __restrict__ X, const float* __restrict__ W,
              const float* __restrict__ BIAS, float* __restrict__ Y, int M) {
    constexpr int K  = F_DIM;
    constexpr int NT = (NC + 15) / 16;

    int wave = threadIdx.x >> 5;
    int mt = blockIdx.x * 4 + wave;          // wave-uniform
    if (mt >= M / 16) return;                // whole-wave exit: EXEC stays all-1s

    int lane = threadIdx.x & 31;
    int hi = lane >> 4;
    int lo = lane & 15;

    const float* xrow = X + (long long)(mt * 16 + lo) * K + 2 * hi;

    v8f acc[NT];
#pragma unroll
    for (int t = 0; t < NT; ++t) acc[t] = (v8f){0, 0, 0, 0, 0, 0, 0, 0};

    for (int kb = 0; kb < K; kb += 4) {
        v2f a;
        a.x = xrow[kb + 0];
        a.y = xrow[kb + 1];
#pragma unroll
        for (int t = 0; t < NT; ++t) {
            int col = t * 16 + lo;
            v2f b;
            if (col < NC) {                  // predicated loads only; WMMA below
                b.x = W[(long long)col * K + kb + 2 * hi + 0];
                b.y = W[(long long)col * K + kb + 2 * hi + 1];
            } else {
                b.x = 0.0f; b.y = 0.0f;
            }
            acc[t] = __builtin_amdgcn_wmma_f32_16x16x4_f32(
                false, a, false, b, (short)0, acc[t], false, false);
        }
    }

#pragma unroll
    for (int t = 0; t < NT; ++t) {
        int col = t * 16 + lo;
        if (col < NC) {
            float bb = BIAS[col];
#pragma unroll
            for (int r = 0; r < 8; ++r) {
                int m = mt * 16 + r + 8 * hi;
                float v = acc[t][r] + bb;
                if (RELU) v = fmaxf(v, 0.0f);
                Y[(long long)m * NC + col] = v;
            }
        }
    }
}

// ---------------------------------------------------------------------------

extern "C" void kernel_launch(void* const* d_in, const int* in_sizes, int n_in,
                              void* d_out, int out_size, void* d_ws, size_t ws_size,
                              hipStream_t stream) {
    const float* X  = (const float*)d_in[0];   // (N, 128)
    const int*   EI = (const int*)d_in[1];     // (2, E)
    const float* W1 = (const float*)d_in[2];   // (128, 128)
    const float* B1 = (const float*)d_in[3];   // (128,)
    const float* W2 = (const float*)d_in[4];   // (40, 128)
    const float* B2 = (const float*)d_in[5];   // (40,)
    float* OUT = (float*)d_out;                // (N, 40)

    const int N = in_sizes[0] / F_DIM;         // 50000 (divisible by 16)
    const int E = in_sizes[1] / 2;             // 800000
    const int* src = EI;
    const int* dst = EI + E;

    // Workspace layout (~51.5 MB): deg/dinv | agg (reused layer2) | h
    char* ws = (char*)d_ws;
    float* deg = (float*)ws;                                         // N floats
    float* agg = (float*)(ws + (size_t)262144);                      // N*128 floats
    float* h   = (float*)(ws + (size_t)262144 + (size_t)N * F_DIM * 4);

    const int T = 256;
    const long long selfT = (long long)N * (F_DIM / 4);
    const long long scatT = (long long)E * F_DIM;
    const int gemmBlocks = (N / 16 + 3) / 4;

    // degree + norm
    sgc_init_deg<<<(N + T - 1) / T, T, 0, stream>>>(deg, N);
    sgc_edge_deg<<<(E + T - 1) / T, T, 0, stream>>>(dst, deg, E);
    sgc_rsqrt<<<(N + T - 1) / T, T, 0, stream>>>(deg, N);

    // layer 1 propagation + GEMM(+ReLU)
    sgc_self_init<<<(int)((selfT + T - 1) / T), T, 0, stream>>>(X, deg, agg, N);
    sgc_scatter<<<(int)((scatT + T - 1) / T), T, 0, stream>>>(src, dst, deg, X, agg, E);
    sgc_gemm_wmma<F_DIM, true><<<gemmBlocks, 128, 0, stream>>>(agg, W1, B1, h, N);

    // layer 2 propagation (reuse agg) + GEMM
    sgc_self_init<<<(int)((selfT + T - 1) / T), T, 0, stream>>>(h, deg, agg, N);
    sgc_scatter<<<(int)((scatT + T - 1) / T), T, 0, stream>>>(src, dst, deg, h, agg, E);
    sgc_gemm_wmma<40, false><<<gemmBlocks, 128, 0, stream>>>(agg, W2, B2, OUT, N);
}